// GCN_net_18537078849729
// MI455X (gfx1250) — compile-verified
//
#include <hip/hip_runtime.h>
#include <hip/hip_bf16.h>

#define N_NODES 10000
#define N_EDGES 640000
#define DIM 128
#define M_TILES (N_NODES / 16)   // 625

typedef __attribute__((ext_vector_type(2))) float v2f;
typedef __attribute__((ext_vector_type(8))) float v8f;

// Native CDNA5 device-scope f32 atomic add (non-returning, STOREcnt-tracked).
// Forces GLOBAL_ATOMIC_ADD_F32 with SCOPE_DEV instead of a possible CAS loop.
__device__ __forceinline__ void atomic_add_f32_dev(float* addr, float val) {
    asm volatile("global_atomic_add_f32 %0, %1, off scope:SCOPE_DEV"
                 :
                 : "v"(addr), "v"(val)
                 : "memory");
}

// ---------------- init kernels ----------------
__global__ void init_deg_kernel(float* __restrict__ deg) {
    int i = blockIdx.x * blockDim.x + threadIdx.x;
    if (i < N_NODES) deg[i] = 1.0f;   // self-loop contribution
}

__global__ void zero_out_kernel(float* __restrict__ out) {
    int i = blockIdx.x * blockDim.x + threadIdx.x;
    if (i < N_NODES * DIM) out[i] = 0.0f;
}

// ---------------- degree (scatter ones onto dst) ----------------
__global__ void deg_kernel(const int* __restrict__ dst, float* __restrict__ deg) {
    int e = blockIdx.x * blockDim.x + threadIdx.x;
    if (e < N_EDGES) {
        int d = dst[e];
        atomic_add_f32_dev(&deg[d], 1.0f);
    }
}

__global__ void dinv_kernel(float* __restrict__ deg) {
    int i = blockIdx.x * blockDim.x + threadIdx.x;
    if (i < N_NODES) deg[i] = rsqrtf(deg[i]);   // deg >= 1 always (self-loop)
}

// ---------------- dense GEMM: xw = x @ W via V_WMMA_F32_16X16X4_F32 --------
// Block = 128 threads (4 waves). Each wave computes one 16-row M-tile across
// all 8 N-tiles (full 128-wide output strip). W staged in LDS (64 KB).
__launch_bounds__(128)
__global__ void gemm_kernel(const float* __restrict__ x,
                            const float* __restrict__ W,
                            float* __restrict__ xw) {
    __shared__ float Ws[DIM * DIM];   // 64 KB, row-major W[k][c]

    // cooperative load of W into LDS, float4-vectorized, coalesced
    {
        const float4* Wg = (const float4*)W;
        float4* Wl = (float4*)Ws;
        for (int i = threadIdx.x; i < DIM * DIM / 4; i += blockDim.x)
            Wl[i] = Wg[i];
    }
    __syncthreads();

    const int wave = threadIdx.x >> 5;
    const int lane = threadIdx.x & 31;
    const int half = lane >> 4;     // 0: lanes 0-15, 1: lanes 16-31
    const int lm   = lane & 15;

    const int mtile = blockIdx.x * 4 + wave;
    if (mtile >= M_TILES) return;   // wave-uniform guard; EXEC stays full inside

    const float* xrow = x + (mtile * 16 + lm) * DIM;  // A row for this lane

    v8f acc[8];
    for (int nt = 0; nt < 8; ++nt)
        for (int r = 0; r < 8; ++r)
            acc[nt][r] = 0.0f;

    for (int kk = 0; kk < 32; ++kk) {
        const int k0 = kk * 4 + half * 2;
        // A-frag: lane holds x[row][k0], x[row][k0+1]  (8B aligned)
        v2f a = *(const v2f*)(xrow + k0);
        const float* wk0 = Ws + k0 * DIM;
        for (int nt = 0; nt < 8; ++nt) {
            const int col = nt * 16 + lm;
            v2f b;
            b.x = wk0[col];         // W[k0  ][col]
            b.y = wk0[DIM + col];   // W[k0+1][col]
            acc[nt] = __builtin_amdgcn_wmma_f32_16x16x4_f32(
                false, a, false, b, (short)0, acc[nt], false, false);
        }
    }

    // store: D lane layout -> rows r + 8*half, col lm within each N-tile
    for (int nt = 0; nt < 8; ++nt) {
        const int col = nt * 16 + lm;
        for (int r = 0; r < 8; ++r) {
            const int row = mtile * 16 + r + 8 * half;
            xw[row * DIM + col] = acc[nt][r];
        }
    }
}

// ---------------- edge scatter: one wave32 per edge, float4 per lane -------
__launch_bounds__(256)
__global__ void scatter_kernel(const int* __restrict__ src,
                               const int* __restrict__ dst,
                               const float* __restrict__ dinv,
                               const float* __restrict__ xw,
                               float* __restrict__ out) {
    const int gid  = blockIdx.x * blockDim.x + threadIdx.x;
    const int e    = gid >> 5;
    const int lane = gid & 31;
    if (e >= N_EDGES) return;

    const int s = src[e];
    const int d = dst[e];
    const float w = dinv[s] * dinv[d];

    const float4 v = ((const float4*)(xw + s * DIM))[lane];  // coalesced b128
    float* o = out + d * DIM + lane * 4;
    atomic_add_f32_dev(o + 0, v.x * w);
    atomic_add_f32_dev(o + 1, v.y * w);
    atomic_add_f32_dev(o + 2, v.z * w);
    atomic_add_f32_dev(o + 3, v.w * w);
}

// ---------------- finalize: self-loop term + bias + ReLU -------------------
__global__ void finalize_kernel(const float* __restrict__ xw,
                                const float* __restrict__ dinv,
                                const float* __restrict__ b,
                                float* __restrict__ out) {
    int idx = blockIdx.x * blockDim.x + threadIdx.x;
    if (idx >= N_NODES * DIM) return;
    const int i = idx >> 7;    // node
    const int j = idx & 127;   // feature
    const float di = dinv[i];
    const float v = out[idx] + xw[idx] * (di * di) + b[j];
    out[idx] = fmaxf(v, 0.0f);
}

extern "C" void kernel_launch(void* const* d_in, const int* in_sizes, int n_in,
                              void* d_out, int out_size, void* d_ws, size_t ws_size,
                              hipStream_t stream) {
    const float* x    = (const float*)d_in[0];
    const int*   edge = (const int*)d_in[1];     // [2, E] flat: src then dst
    const float* W    = (const float*)d_in[2];
    const float* b    = (const float*)d_in[3];
    float* out = (float*)d_out;

    const int* src = edge;
    const int* dst = edge + N_EDGES;

    // workspace: dinv[N] (padded to 10240) then xw[N*DIM]  (~5.16 MB total)
    float* dinv = (float*)d_ws;
    float* xw   = dinv + 10240;

    init_deg_kernel<<<(N_NODES + 255) / 256, 256, 0, stream>>>(dinv);
    zero_out_kernel<<<(N_NODES * DIM + 255) / 256, 256, 0, stream>>>(out);
    deg_kernel<<<(N_EDGES + 255) / 256, 256, 0, stream>>>(dst, dinv);
    dinv_kernel<<<(N_NODES + 255) / 256, 256, 0, stream>>>(dinv);

    gemm_kernel<<<(M_TILES + 3) / 4, 128, 0, stream>>>(x, W, xw);

    // one wave per edge: E * 32 threads
    const long long sthreads = (long long)N_EDGES * 32;
    scatter_kernel<<<(int)((sthreads + 255) / 256), 256, 0, stream>>>(src, dst, dinv, xw, out);

    finalize_kernel<<<(N_NODES * DIM + 255) / 256, 256, 0, stream>>>(xw, dinv, b, out);
}